// MeshNodeBlock_57552561766959
// MI455X (gfx1250) — compile-verified
//
#include <hip/hip_runtime.h>

// CDNA5 / gfx1250 fused MeshNodeBlock:
//   agg = segment_sum(edge_features, src)               (scatter, hw f32 atomics)
//   out = LN(silu([node|agg]@W1+b1)@W2+b2)*g+b + node   (fused fp32-WMMA kernel)

typedef __attribute__((ext_vector_type(2))) float v2f;
typedef __attribute__((ext_vector_type(8))) float v8f;

#define D_NODE 128
#define D_IN   256
#define D_HID  512
#define D_OUT  128
#define BM      32            // node rows per block
#define NTHREADS 256          // 8 waves
#define XS_STRIDE 257         // 256 + 1 pad -> conflict-free A-frag reads
#define HS_STRIDE 513         // 512 + 1 pad

__device__ __forceinline__ v8f wmma_f32(v2f a, v2f b, v8f c) {
  // D = A(16x4) * B(4x16) + C(16x16), full fp32 on the matrix pipe
  return __builtin_amdgcn_wmma_f32_16x16x4_f32(false, a, false, b, (short)0, c,
                                               false, false);
}

__device__ __forceinline__ float silu(float v) {
  // v * sigmoid(v); v_rcp_f32 (1 ULP) instead of IEEE divide sequence
  return v * __builtin_amdgcn_rcpf(1.f + __expf(-v));
}

__global__ void zero_kernel(float4* __restrict__ p, int n4) {
  int i = blockIdx.x * blockDim.x + threadIdx.x;
  if (i < n4) p[i] = make_float4(0.f, 0.f, 0.f, 0.f);
}

__global__ void scatter_kernel(const float* __restrict__ edge,
                               const int* __restrict__ src,
                               float* __restrict__ agg, long total) {
  long gid = (long)blockIdx.x * blockDim.x + threadIdx.x;
  if (gid >= total) return;
  int e  = (int)(gid >> 5);
  int c4 = ((int)gid & 31) << 2;           // float4 column
  int n  = src[e];
  const float4 v = *(const float4*)(edge + (size_t)e * D_NODE + c4);
  float* a = agg + (size_t)n * D_NODE + c4;
  unsafeAtomicAdd(a + 0, v.x);
  unsafeAtomicAdd(a + 1, v.y);
  unsafeAtomicAdd(a + 2, v.z);
  unsafeAtomicAdd(a + 3, v.w);
}

__launch_bounds__(NTHREADS)
__global__ void fused_kernel(const float* __restrict__ node,
                             const float* __restrict__ agg,
                             const float* __restrict__ W1,
                             const float* __restrict__ b1,
                             const float* __restrict__ W2,
                             const float* __restrict__ b2,
                             const float* __restrict__ gamma,
                             const float* __restrict__ beta,
                             float* __restrict__ out) {
  __shared__ __align__(16) float xs[BM * XS_STRIDE];  // x tile; reused as ys
  __shared__ __align__(16) float hs[BM * HS_STRIDE];  // hidden tile
  float* ys = xs;                                     // [BM][128], xs dead by then

  const int tid  = threadIdx.x;
  const int wave = tid >> 5;           // 8 waves
  const int lane = tid & 31;
  const int half = lane >> 4;          // 0: lanes 0-15, 1: lanes 16-31
  const int lm   = lane & 15;
  const int row0 = blockIdx.x * BM;

  // ---- stage x = [node | agg] into LDS (coalesced float4) ----
  for (int i = tid; i < BM * (D_IN / 4); i += NTHREADS) {
    int r  = i >> 6;                   // D_IN/4 == 64 float4s per row
    int c4 = (i & 63) << 2;
    float4 v;
    if (c4 < D_NODE)
      v = *(const float4*)(node + (size_t)(row0 + r) * D_NODE + c4);
    else
      v = *(const float4*)(agg + (size_t)(row0 + r) * D_NODE + (c4 - D_NODE));
    float* d = xs + r * XS_STRIDE + c4;
    d[0] = v.x; d[1] = v.y; d[2] = v.z; d[3] = v.w;
  }
  __syncthreads();

  // ---- GEMM1: h = silu(x @ W1 + b1), wave owns 64 hidden cols ----
  {
    const float* xr0 = xs + lm * XS_STRIDE + 2 * half;         // M-tile 0 row
    const float* xr1 = xs + (lm + 16) * XS_STRIDE + 2 * half;  // M-tile 1 row
    for (int nt = 0; nt < 4; ++nt) {
      const int n0 = wave * 64 + nt * 16;
      const float* w1c = W1 + (size_t)(2 * half) * D_HID + n0 + lm;
      v8f acc0 = {};
      v8f acc1 = {};
#pragma unroll 4
      for (int k0 = 0; k0 < D_IN; k0 += 4) {
        v2f a0 = { xr0[k0], xr0[k0 + 1] };
        v2f a1 = { xr1[k0], xr1[k0 + 1] };
        v2f b  = { w1c[(size_t)k0 * D_HID], w1c[(size_t)(k0 + 1) * D_HID] };
        acc0 = wmma_f32(a0, b, acc0);
        acc1 = wmma_f32(a1, b, acc1);
      }
      const float bias = b1[n0 + lm];
      float* h0 = hs + (half * 8) * HS_STRIDE + n0 + lm;
      float* h1 = hs + (16 + half * 8) * HS_STRIDE + n0 + lm;
#pragma unroll
      for (int j = 0; j < 8; ++j)
        h0[j * HS_STRIDE] = silu(acc0[j] + bias);
#pragma unroll
      for (int j = 0; j < 8; ++j)
        h1[j * HS_STRIDE] = silu(acc1[j] + bias);
    }
  }
  __syncthreads();

  // ---- GEMM2: y = h @ W2 + b2, wave owns 16 out cols ----
  {
    const float* hr0 = hs + lm * HS_STRIDE + 2 * half;
    const float* hr1 = hs + (lm + 16) * HS_STRIDE + 2 * half;
    const int n0 = wave * 16;
    const float* w2c = W2 + (size_t)(2 * half) * D_OUT + n0 + lm;
    v8f acc0 = {};
    v8f acc1 = {};
#pragma unroll 4
    for (int k0 = 0; k0 < D_HID; k0 += 4) {
      v2f a0 = { hr0[k0], hr0[k0 + 1] };
      v2f a1 = { hr1[k0], hr1[k0 + 1] };
      v2f b  = { w2c[(size_t)k0 * D_OUT], w2c[(size_t)(k0 + 1) * D_OUT] };
      acc0 = wmma_f32(a0, b, acc0);
      acc1 = wmma_f32(a1, b, acc1);
    }
    const float bias = b2[n0 + lm];
#pragma unroll
    for (int j = 0; j < 8; ++j) {
      ys[(half * 8 + j) * D_OUT + n0 + lm]      = acc0[j] + bias;
      ys[(16 + half * 8 + j) * D_OUT + n0 + lm] = acc1[j] + bias;
    }
  }
  __syncthreads();

  // ---- LayerNorm + residual: wave handles 4 rows, float4/lane, shfl reduce ----
  {
    const float4 g  = *(const float4*)(gamma + lane * 4);
    const float4 be = *(const float4*)(beta + lane * 4);
    for (int i = 0; i < 4; ++i) {
      const int r = wave * 4 + i;
      float4 v = *(const float4*)(ys + r * D_OUT + lane * 4);
      float s  = v.x + v.y + v.z + v.w;
      float sq = v.x * v.x + v.y * v.y + v.z * v.z + v.w * v.w;
#pragma unroll
      for (int off = 16; off >= 1; off >>= 1) {
        s  += __shfl_xor(s, off, 32);
        sq += __shfl_xor(sq, off, 32);
      }
      const float mu  = s * (1.f / D_OUT);
      const float var = sq * (1.f / D_OUT) - mu * mu;
      const float rs  = __builtin_amdgcn_rsqf(var + 1e-5f);
      const float4 nf = *(const float4*)(node + (size_t)(row0 + r) * D_NODE + lane * 4);
      float4 o;
      o.x = (v.x - mu) * rs * g.x + be.x + nf.x;
      o.y = (v.y - mu) * rs * g.y + be.y + nf.y;
      o.z = (v.z - mu) * rs * g.z + be.z + nf.z;
      o.w = (v.w - mu) * rs * g.w + be.w + nf.w;
      *(float4*)(out + (size_t)(row0 + r) * D_OUT + lane * 4) = o;
    }
  }
}

extern "C" void kernel_launch(void* const* d_in, const int* in_sizes, int n_in,
                              void* d_out, int out_size, void* d_ws, size_t ws_size,
                              hipStream_t stream) {
  const float* node  = (const float*)d_in[0];
  const float* edge  = (const float*)d_in[1];
  const float* W1    = (const float*)d_in[2];
  const float* b1    = (const float*)d_in[3];
  const float* W2    = (const float*)d_in[4];
  const float* b2    = (const float*)d_in[5];
  const float* gamma = (const float*)d_in[6];
  const float* beta  = (const float*)d_in[7];
  const int*   src   = (const int*)d_in[8];
  float* out = (float*)d_out;
  float* agg = (float*)d_ws;          // [n_nodes][128] scratch, 51.2 MB

  const int n_nodes = in_sizes[0] / D_NODE;   // 100000
  const int n_edges = in_sizes[1] / D_NODE;   // 600000

  const int n4 = n_nodes * (D_NODE / 4);
  zero_kernel<<<(n4 + 255) / 256, 256, 0, stream>>>((float4*)agg, n4);

  const long total = (long)n_edges * 32;      // one float4 per thread
  scatter_kernel<<<(int)((total + 255) / 256), 256, 0, stream>>>(edge, src, agg, total);

  fused_kernel<<<n_nodes / BM, NTHREADS, 0, stream>>>(node, agg, W1, b1, W2, b2,
                                                      gamma, beta, out);
}